// NMGaussianMxiture_2662879723706
// MI455X (gfx1250) — compile-verified
//
#include <hip/hip_runtime.h>
#include <math.h>

typedef float v2f __attribute__((ext_vector_type(2)));
typedef float v8f __attribute__((ext_vector_type(8)));

#define Kc      16
#define Dc      8
#define NPAIR   256     // K*K
#define NFEAT   44      // 8 diag + 28 upper-tri (x2 folded) + 8 linear
#define KSTEPS  11      // 44 / 4 (K per V_WMMA_F32_16X16X4_F32)
#define CSTRIDE 272     // LDS row stride for C (16*17: halves hit disjoint banks)
#define FSTRIDE 52      // per-point feature stride (52m mod 64 distinct, m<16)

// workspace float layout
#define WS_ACC    0
#define WS_ZLAST  1
#define WS_COEF   2
#define WS_MCONST (2 + NPAIR)
#define WS_CMAT   (2 + 2 * NPAIR)
#define WS_TOTAL  (WS_CMAT + NFEAT * NPAIR)

// Gauss-Jordan 8x8 inverse (SPD inputs, no pivoting needed); returns det(A).
__device__ float invert8(const float* A, float* Ainv) {
    float m[8][16];
    for (int r = 0; r < 8; ++r)
        for (int c = 0; c < 8; ++c) {
            m[r][c]     = A[r * 8 + c];
            m[r][c + 8] = (r == c) ? 1.0f : 0.0f;
        }
    float det = 1.0f;
    for (int p = 0; p < 8; ++p) {
        float piv = m[p][p];
        det *= piv;
        float ip = 1.0f / piv;
        for (int c = 0; c < 16; ++c) m[p][c] *= ip;
        for (int r = 0; r < 8; ++r) {
            if (r == p) continue;
            float f = m[r][p];
            for (int c = 0; c < 16; ++c) m[r][c] -= f * m[p][c];
        }
    }
    for (int r = 0; r < 8; ++r)
        for (int c = 0; c < 8; ++c) Ainv[r * 8 + c] = m[r][c + 8];
    return det;
}

// One block, 256 threads: per-component precisions (16 threads), then one
// thread per (i,j) pair builds the fused GEMM coefficients.
__global__ void setup_kernel(const float* __restrict__ mu,
                             const float* __restrict__ L,
                             const float* __restrict__ w,
                             float* ws) {
    __shared__ float sPrec[Kc][64];
    __shared__ float sRhs[Kc][Dc];
    int t = threadIdx.x;
    if (t == 0) ws[WS_ACC] = 0.0f;

    if (t < Kc) {
        // sig = tril(L) @ tril(L)^T + I
        float sig[64];
        for (int r = 0; r < 8; ++r)
            for (int c = 0; c < 8; ++c) {
                float s = (r == c) ? 1.0f : 0.0f;
                int mn = (r < c) ? r : c;
                for (int e = 0; e <= mn; ++e)
                    s += L[t * 64 + r * 8 + e] * L[t * 64 + c * 8 + e];
                sig[r * 8 + c] = s;
            }
        float inv[64];
        invert8(sig, inv);
        for (int q = 0; q < 64; ++q) sPrec[t][q] = inv[q];
        for (int d = 0; d < 8; ++d) {
            float s = 0.0f;
            for (int e = 0; e < 8; ++e) s += inv[d * 8 + e] * mu[t * 8 + e];
            sRhs[t][d] = s;
        }
    }
    __syncthreads();

    int p = t;                 // 256 pairs, one per thread
    int i = p >> 4, j = p & 15;

    float P[64], Sig[64];
    for (int q = 0; q < 64; ++q) P[q] = sPrec[i][q] + sPrec[j][q];
    float detP = invert8(P, Sig);
    float detS = 1.0f / detP;  // det(inv(P))

    float rp[8], mp[8], b[8];
    for (int d = 0; d < 8; ++d) rp[d] = sRhs[i][d] + sRhs[j][d];
    for (int d = 0; d < 8; ++d) {
        float s = 0.0f;
        for (int e = 0; e < 8; ++e) s += Sig[d * 8 + e] * rp[e];
        mp[d] = s;
    }
    float muPmu = 0.0f;
    for (int d = 0; d < 8; ++d) {
        float s = 0.0f;
        for (int e = 0; e < 8; ++e) s += P[d * 8 + e] * mp[e];
        b[d] = s;
        muPmu += mp[d] * s;
    }

    // Faithful to the reference's f32 semantics: (2*pi)^64 overflows float32
    // (JAX weak-typed scalar -> inf), giving z == 0 exactly as jnp does.
    float tp = 1.0f;
    for (int q = 0; q < 64; ++q) tp *= 6.28318530717958647692f;
    float z = 1.0f / sqrtf(tp * detS);

    ws[WS_COEF + p]   = z * w[i] * w[j];
    ws[WS_MCONST + p] = -0.5f * muPmu;

    // Symmetric fold of x'Px: diag gets -0.5*P[d][d]; off-diag pair (d<e)
    // gets -0.5*(P[d][e]+P[e][d]) (sum of both reference terms, exact).
    for (int d = 0; d < 8; ++d)
        ws[WS_CMAT + d * NPAIR + p] = -0.5f * P[d * 8 + d];
    int f = 8;
    for (int d = 0; d < 8; ++d)
        for (int e = d + 1; e < 8; ++e)
            ws[WS_CMAT + (f++) * NPAIR + p] = -0.5f * (P[d * 8 + e] + P[e * 8 + d]);
    for (int d = 0; d < 8; ++d)
        ws[WS_CMAT + (36 + d) * NPAIR + p] = b[d];

    if (p == NPAIR - 1) ws[WS_ZLAST] = z;
}

// Main kernel: each wave owns a 16-point tile and sweeps all 256 pair columns
// with V_WMMA_F32_16X16X4_F32.  C[44,256] and per-pair scalars live in LDS.
__global__ void __launch_bounds__(256)
mix_kernel(const float* __restrict__ X, float* ws, int N) {
    extern __shared__ float lds[];
    float* sCmat   = lds;                       // NFEAT * CSTRIDE
    float* sCoef   = sCmat + NFEAT * CSTRIDE;   // NPAIR
    float* sMconst = sCoef + NPAIR;             // NPAIR
    float* sFeat   = sMconst + NPAIR;           // 8 waves * 16 pts * FSTRIDE

    // Stage coefficient matrix + scalars into LDS (whole block cooperates).
    for (int idx = threadIdx.x; idx < NFEAT * NPAIR; idx += blockDim.x) {
        int r = idx >> 8, c = idx & 255;
        sCmat[r * CSTRIDE + c] = ws[WS_CMAT + idx];
    }
    for (int idx = threadIdx.x; idx < NPAIR; idx += blockDim.x) {
        sCoef[idx]   = ws[WS_COEF + idx];
        sMconst[idx] = ws[WS_MCONST + idx];
    }

    int wave = threadIdx.x >> 5;
    int lane = threadIdx.x & 31;
    int m    = lane & 15;       // row within tile (also column-within-tile id)
    int half = lane >> 4;       // K-half for A/B fragments
    long tileBase = ((long)blockIdx.x * 8 + wave) * 16;
    bool active = tileBase < (long)N;

    if (active) {
        long gp = tileBase + m;
        if (gp > (long)N - 1) gp = (long)N - 1;   // clamp tail (masked later)
        const float* xr = X + gp * Dc;
        float x[8];
        #pragma unroll
        for (int d = 0; d < 8; ++d) x[d] = xr[d];

        // Both halves write identical values to identical addresses: harmless
        // duplicate store, keeps all indices compile-time constant.
        float* fr = sFeat + (wave * 16 + m) * FSTRIDE;
        #pragma unroll
        for (int d = 0; d < 8; ++d) fr[d] = x[d] * x[d];
        {
            int f = 8;
            #pragma unroll
            for (int d = 0; d < 8; ++d)
                #pragma unroll
                for (int e = d + 1; e < 8; ++e)
                    fr[f++] = x[d] * x[e];
        }
        #pragma unroll
        for (int d = 0; d < 8; ++d) fr[36 + d] = x[d];
    }
    __syncthreads();

    float zlast = ws[WS_ZLAST];
    if (active) {
        // Preload all A fragments once; reuse across 16 column tiles.
        // A 16x4 f32 layout: lanes 0-15 hold K={0,1}, lanes 16-31 K={2,3}.
        float areg[2 * KSTEPS];
        const float* fr = sFeat + (wave * 16 + m) * FSTRIDE;
        #pragma unroll
        for (int kk = 0; kk < KSTEPS; ++kk) {
            areg[2 * kk + 0] = fr[kk * 4 + half * 2 + 0];
            areg[2 * kk + 1] = fr[kk * 4 + half * 2 + 1];
        }

        float acc[8];
        #pragma unroll
        for (int v = 0; v < 8; ++v) acc[v] = 0.0f;

        for (int j = 0; j < 16; ++j) {
            int col = j * 16 + m;
            float mc = sMconst[col];          // mu'P mu term folded into C init
            v8f c;
            #pragma unroll
            for (int v = 0; v < 8; ++v) c[v] = mc;

            #pragma unroll
            for (int kk = 0; kk < KSTEPS; ++kk) {
                v2f a; a[0] = areg[2 * kk]; a[1] = areg[2 * kk + 1];
                v2f b;
                const float* brow = sCmat + (kk * 4 + half * 2) * CSTRIDE + col;
                b[0] = brow[0];
                b[1] = brow[CSTRIDE];
                c = __builtin_amdgcn_wmma_f32_16x16x4_f32(
                        false, a, false, b, (short)0, c, false, false);
            }
            float coef = sCoef[col];
            #pragma unroll
            for (int v = 0; v < 8; ++v) acc[v] += __expf(c[v]) * coef;
        }

        // acc[v] holds a 16-column partial of pdf for row (half*8 + v);
        // reduce across the 16 lanes of each half.
        #pragma unroll
        for (int v = 0; v < 8; ++v) {
            #pragma unroll
            for (int off = 1; off < 16; off <<= 1)
                acc[v] += __shfl_xor(acc[v], off, 32);
        }

        float lsum = 0.0f;
        if (m == 0) {
            #pragma unroll
            for (int v = 0; v < 8; ++v) {
                long gp = tileBase + half * 8 + v;
                if (gp < (long)N) lsum += logf(zlast * acc[v]);
            }
        }
        lsum += __shfl_xor(lsum, 16, 32);
        if (lane == 0) atomicAdd(&ws[WS_ACC], lsum);
    }
}

__global__ void finalize_kernel(const float* ws, float* out, int N) {
    out[0] = ws[WS_ACC] / (float)N;
}

extern "C" void kernel_launch(void* const* d_in, const int* in_sizes, int n_in,
                              void* d_out, int out_size, void* d_ws, size_t ws_size,
                              hipStream_t stream) {
    const float* X  = (const float*)d_in[0];
    const float* mu = (const float*)d_in[1];
    const float* L  = (const float*)d_in[2];
    const float* w  = (const float*)d_in[3];
    int N = in_sizes[0] / Dc;
    float* ws = (float*)d_ws;

    size_t ldsBytes =
        (size_t)(NFEAT * CSTRIDE + 2 * NPAIR + 8 * 16 * FSTRIDE) * sizeof(float);
    // Allow >64KB dynamic LDS if needed (gfx1250 WGP has 320KB); ignore errors.
    (void)hipFuncSetAttribute((const void*)mix_kernel,
                              hipFuncAttributeMaxDynamicSharedMemorySize,
                              (int)ldsBytes);

    setup_kernel<<<1, 256, 0, stream>>>(mu, L, w, ws);

    int nTiles  = (N + 15) / 16;
    int nBlocks = (nTiles + 7) / 8;
    mix_kernel<<<nBlocks, 256, ldsBytes, stream>>>(X, ws, N);

    finalize_kernel<<<1, 1, 0, stream>>>(ws, (float*)d_out, N);
}